// SelfAttentionBlock_35923106463789
// MI455X (gfx1250) — compile-verified
//
#include <hip/hip_runtime.h>
#include <math.h>

// ---------------------------------------------------------------------------
// Self-attention: out = softmax((xWq)(xWk)^T / sqrt(d)) (xWv)
// N = 4096, D_IN = 1024, UNITS = 1024, fp32 end to end.
//
// All GEMMs run as C = A @ Bt^T with Bt stored [N,K] row-major (small B
// operands are pre-transposed by a tiled transpose kernel). This makes every
// WMMA B-fragment a contiguous {k,k+1} pair in LDS -> single aligned
// ds_load_b64 directly into an even VGPR pair (no repacking movs).
//
// Compute: V_WMMA_F32_16X16X4_F32 (exact fp32 vs the reference).
// Data movement: double-buffered LDS fed by GLOBAL_LOAD_ASYNC_TO_LDS_B128
// (ASYNCcnt) so the async engines fill tile t+1 while WMMAs consume tile t.
// ---------------------------------------------------------------------------

typedef float v2f __attribute__((ext_vector_type(2)));
typedef float v8f __attribute__((ext_vector_type(8)));
typedef int   v4i __attribute__((ext_vector_type(4)));

#define TILE_M 64
#define TILE_N 128
#define TILE_K 32
#define LDS_STRIDE 36   // (row, k) stride in floats: lane addrs distinct mod 64 banks,
                        // 36*4 % 16 == 0 so 16B async chunks stay aligned

#if __has_builtin(__builtin_amdgcn_global_load_async_to_lds_b128) && \
    __has_builtin(__builtin_amdgcn_s_wait_asynccnt)
#define ASYNC_LDS 1
#else
#define ASYNC_LDS 0
#endif

// Copy 16 aligned bytes global -> LDS. Async path: no VGPR round trip,
// tracked by ASYNCcnt. Fallback: plain load + ds_store.
__device__ __forceinline__ void copy16_to_lds(const float* g, float* l) {
#if ASYNC_LDS
  __builtin_amdgcn_global_load_async_to_lds_b128(
      (__attribute__((address_space(1))) v4i*)(g),
      (__attribute__((address_space(3))) v4i*)(l), 0, 0);
#else
  *(float4*)l = *(const float4*)g;
#endif
}

// 12 async ops per wave per K-tile; async completion is in-order per wave,
// so waiting for <=12 outstanding guarantees the previous tile has landed.
__device__ __forceinline__ void wait_stage_keep12() {
#if ASYNC_LDS
  __builtin_amdgcn_s_wait_asynccnt(12);
#endif
}
__device__ __forceinline__ void wait_stage_all() {
#if ASYNC_LDS
  __builtin_amdgcn_s_wait_asynccnt(0);
#endif
}

// Issue the 12 16-byte copies this thread owns for one (A, Bt) K-tile.
// A tile:  64 rows (m) x 32 (k), LDS layout (m, k) stride 36.
// Bt tile: 128 rows (n) x 32 (k), LDS layout (n, k) stride 36.
__device__ __forceinline__ void stage_tile(const float* __restrict__ A,
                                           const float* __restrict__ Bt,
                                           int lda, int ldb, int rowBase,
                                           int colBase, int kb, float* as,
                                           float* bs, int tid) {
#pragma unroll
  for (int i = 0; i < 4; ++i) {            // 512 float4, 4 per thread (128 thr)
    int f = tid + i * 128;
    int r = f >> 3;                        // 0..63
    int c4 = (f & 7) << 2;                 // 0,4,..,28
    copy16_to_lds(A + (size_t)(rowBase + r) * lda + kb + c4,
                  as + r * LDS_STRIDE + c4);
  }
#pragma unroll
  for (int i = 0; i < 8; ++i) {            // 1024 float4, 8 per thread
    int f = tid + i * 128;
    int n = f >> 3;                        // 0..127
    int k4 = (f & 7) << 2;                 // 0,4,..,28
    copy16_to_lds(Bt + (size_t)(colBase + n) * ldb + kb + k4,
                  bs + n * LDS_STRIDE + k4);
  }
}

// C[M,N] = alpha * A[M,K] @ Bt[N,K]^T.  Grid: (N/TILE_N, M/TILE_M).
// Block: 128 threads = 4 wave32s in a 2(M) x 2(N) grid; each wave owns a
// 32x64 tile = 2x4 WMMA accumulators (64 VGPRs). All dims tile-exact.
__global__ __launch_bounds__(128)
void wmma_gemm_nt_f32(const float* __restrict__ A, const float* __restrict__ Bt,
                      float* __restrict__ C, int lda, int ldb, int ldc,
                      int K, float alpha) {
  __shared__ float As[2][TILE_M * LDS_STRIDE];   // 2 x 9216 B
  __shared__ float Bs[2][TILE_N * LDS_STRIDE];   // 2 x 18432 B  (54 KB total)

  const int tid  = threadIdx.x;
  const int lane = tid & 31;
  const int wave = tid >> 5;     // 0..3
  const int lr   = lane & 15;    // lane within half-wave
  const int hi   = lane >> 4;    // 0 = lanes 0-15, 1 = lanes 16-31

  const int m0 = (wave & 1) * 32;        // wave tile origin: 32 rows
  const int n0 = (wave >> 1) * 64;       // 64 cols

  const int rowBase = blockIdx.y * TILE_M;
  const int colBase = blockIdx.x * TILE_N;

  v8f acc[2][4] = {};   // 2(M) x 4(N) grid of 16x16 f32 accumulators

  const int nTiles = K / TILE_K;

  // Prologue: start filling buffer 0.
  stage_tile(A, Bt, lda, ldb, rowBase, colBase, 0, As[0], Bs[0], tid);

  for (int t = 0; t < nTiles; ++t) {
    const int cur = t & 1;

    // Kick off tile t+1 into the other buffer (its previous readers finished
    // at the closing barrier of iteration t-1), then wait for tile t's ops.
    if (t + 1 < nTiles) {
      stage_tile(A, Bt, lda, ldb, rowBase, colBase, (t + 1) * TILE_K,
                 As[cur ^ 1], Bs[cur ^ 1], tid);
      wait_stage_keep12();
    } else {
      wait_stage_all();
    }
    __syncthreads();   // all waves' copies for `cur` are visible

    // ---- compute on buffer `cur`: 8 k-chunks of 4 => 64 v_wmma ----
#pragma unroll
    for (int kc = 0; kc < TILE_K; kc += 4) {
      v2f a[2], b[4];
      // A fragment (ISA layout): lane lr holds row m0+mt*16+lr,
      // K = kc+2*hi, kc+2*hi+1 -> adjacent floats, one b64 from LDS.
#pragma unroll
      for (int mt = 0; mt < 2; ++mt) {
        const float* p = &As[cur][(m0 + mt * 16 + lr) * LDS_STRIDE + kc + (hi << 1)];
        a[mt].x = p[0];
        a[mt].y = p[1];
      }
      // B fragment: v0 = Bt[col][kc+2*hi], v1 = Bt[col][kc+2*hi+1] -> b64.
#pragma unroll
      for (int nt = 0; nt < 4; ++nt) {
        const float* p = &Bs[cur][(n0 + nt * 16 + lr) * LDS_STRIDE + kc + (hi << 1)];
        b[nt].x = p[0];
        b[nt].y = p[1];
      }
#pragma unroll
      for (int mt = 0; mt < 2; ++mt)
#pragma unroll
        for (int nt = 0; nt < 4; ++nt)
          acc[mt][nt] = __builtin_amdgcn_wmma_f32_16x16x4_f32(
              false, a[mt], false, b[nt], (short)0, acc[mt][nt], false, false);
    }
    __syncthreads();   // everyone done reading `cur` before it is refilled
  }

  // ---- epilogue: C/D layout: VGPR r -> row (m + r + 8*hi), col = n + lr ----
#pragma unroll
  for (int mt = 0; mt < 2; ++mt) {
#pragma unroll
    for (int nt = 0; nt < 4; ++nt) {
      const int col = colBase + n0 + nt * 16 + lr;
#pragma unroll
      for (int r = 0; r < 8; ++r) {
        const int row = rowBase + m0 + mt * 16 + (hi << 3) + r;
        C[(size_t)row * ldc + col] = alpha * acc[mt][nt][r];
      }
    }
  }
}

// ---------------------------------------------------------------------------
// Tiled transpose: out[c][r] = in[r][c].  Grid: (cols/32, rows/32), block 256.
// Coalesced on both sides via a padded LDS tile.
// ---------------------------------------------------------------------------
__global__ __launch_bounds__(256)
void transpose_f32(const float* __restrict__ in, float* __restrict__ out,
                   int rows, int cols) {
  __shared__ float tile[32][33];
  const int bx = blockIdx.x * 32;   // col base (in)
  const int by = blockIdx.y * 32;   // row base (in)
  const int tx = threadIdx.x & 31;
  const int ty = threadIdx.x >> 5;  // 0..7
#pragma unroll
  for (int i = 0; i < 32; i += 8)
    tile[ty + i][tx] = in[(size_t)(by + ty + i) * cols + bx + tx];
  __syncthreads();
#pragma unroll
  for (int i = 0; i < 32; i += 8)
    out[(size_t)(bx + ty + i) * rows + by + tx] = tile[tx][ty + i];
}

// ---------------------------------------------------------------------------
// Row softmax over S[row][0..N-1], in place. One 256-thread block per row.
// ---------------------------------------------------------------------------
__global__ __launch_bounds__(256)
void softmax_rows(float* __restrict__ S, int N) {
  __shared__ float red[8];
  const int tid = threadIdx.x;
  float* p = S + (size_t)blockIdx.x * N;

  float m = -INFINITY;
  for (int c = tid; c < N; c += 256) m = fmaxf(m, p[c]);
#pragma unroll
  for (int off = 16; off > 0; off >>= 1) m = fmaxf(m, __shfl_xor(m, off, 32));
  if ((tid & 31) == 0) red[tid >> 5] = m;
  __syncthreads();
  m = red[0];
#pragma unroll
  for (int i = 1; i < 8; ++i) m = fmaxf(m, red[i]);
  __syncthreads();

  float s = 0.0f;
  for (int c = tid; c < N; c += 256) {
    float e = __expf(p[c] - m);
    p[c] = e;
    s += e;
  }
#pragma unroll
  for (int off = 16; off > 0; off >>= 1) s += __shfl_xor(s, off, 32);
  if ((tid & 31) == 0) red[tid >> 5] = s;
  __syncthreads();
  float tot = 0.0f;
#pragma unroll
  for (int i = 0; i < 8; ++i) tot += red[i];
  const float inv = 1.0f / tot;

  for (int c = tid; c < N; c += 256) p[c] *= inv;
}

// ---------------------------------------------------------------------------
// Harness entry point.
// ---------------------------------------------------------------------------
extern "C" void kernel_launch(void* const* d_in, const int* in_sizes, int n_in,
                              void* d_out, int out_size, void* d_ws, size_t ws_size,
                              hipStream_t stream) {
  (void)in_sizes; (void)n_in; (void)out_size; (void)ws_size;

  const int N = 4096, D = 1024, U = 1024;
  const float scale = 0.03125f;   // 1/sqrt(1024)

  const float* x  = (const float*)d_in[0];
  const float* Wq = (const float*)d_in[1];
  const float* Wk = (const float*)d_in[2];
  const float* Wv = (const float*)d_in[3];
  float* out = (float*)d_out;

  // workspace: Q | K | V | Vt | Wtq | Wtk | Wtv | S
  // 16 + 16 + 16 + 16 + 4 + 4 + 4 + 64 MB = 140 MB
  float* Q   = (float*)d_ws;
  float* Km  = Q   + (size_t)N * U;
  float* V   = Km  + (size_t)N * U;
  float* Vt  = V   + (size_t)N * U;
  float* Wtq = Vt  + (size_t)N * U;
  float* Wtk = Wtq + (size_t)D * U;
  float* Wtv = Wtk + (size_t)D * U;
  float* S   = Wtv + (size_t)D * U;

  dim3 blkT(256), blkG(128), blkS(256);

  // Transpose weights: Wt[u][d] = W[d][u]  (3 x 4 MB, negligible traffic)
  dim3 gWT(U / 32, D / 32);
  transpose_f32<<<gWT, blkT, 0, stream>>>(Wq, Wtq, D, U);
  transpose_f32<<<gWT, blkT, 0, stream>>>(Wk, Wtk, D, U);
  transpose_f32<<<gWT, blkT, 0, stream>>>(Wv, Wtv, D, U);

  // QKV projections: [4096,1024] = x[4096,1024] @ Wt[1024,1024]^T
  dim3 gProj(U / TILE_N, N / TILE_M);
  wmma_gemm_nt_f32<<<gProj, blkG, 0, stream>>>(x, Wtq, Q,  D, D, U, D, 1.0f);
  wmma_gemm_nt_f32<<<gProj, blkG, 0, stream>>>(x, Wtk, Km, D, D, U, D, 1.0f);
  wmma_gemm_nt_f32<<<gProj, blkG, 0, stream>>>(x, Wtv, V,  D, D, U, D, 1.0f);

  // Vt[u][n] = V[n][u] for the final NT GEMM
  dim3 gVT(U / 32, N / 32);
  transpose_f32<<<gVT, blkT, 0, stream>>>(V, Vt, N, U);

  // scores: [4096,4096] = Q @ K^T, scaled (K already [N,K] row-major)
  dim3 gScore(N / TILE_N, N / TILE_M);
  wmma_gemm_nt_f32<<<gScore, blkG, 0, stream>>>(Q, Km, S, U, U, N, U, scale);

  // softmax along rows
  softmax_rows<<<N, blkS, 0, stream>>>(S, N);

  // out: [4096,1024] = P[4096,4096] @ Vt[1024,4096]^T
  dim3 gOut(U / TILE_N, N / TILE_M);
  wmma_gemm_nt_f32<<<gOut, blkG, 0, stream>>>(S, Vt, out, N, N, U, N, 1.0f);
}